// ODEFunc_17497696764526
// MI455X (gfx1250) — compile-verified
//
#include <hip/hip_runtime.h>

typedef __attribute__((ext_vector_type(2))) float v2f;
typedef __attribute__((ext_vector_type(8))) float v8f;

// Hardware fp32 atomic add at device (agent) scope -> global_atomic_add_f32.
__device__ __forceinline__ void fadd_atomic(float* p, float v) {
    __hip_atomic_fetch_add(p, v, __ATOMIC_RELAXED, __HIP_MEMORY_SCOPE_AGENT);
}

// Pass 1: A[row] += w * x[col];  B[row] += w.   4 edges per thread.
__global__ void edge_pass1(const int* __restrict__ rows, const int* __restrict__ cols,
                           const float* __restrict__ w, const float* __restrict__ x,
                           float* __restrict__ A, float* __restrict__ Bs, int E)
{
    int t = blockIdx.x * blockDim.x + threadIdx.x;
    int base = t * 4;
    if (base + 4 <= E) {
        const int4   r4 = *(const int4*)(rows + base);
        const int4   c4 = *(const int4*)(cols + base);
        const float4 w4 = *(const float4*)(w + base);
        float x0 = x[c4.x], x1 = x[c4.y], x2 = x[c4.z], x3 = x[c4.w];
        fadd_atomic(&A[r4.x], w4.x * x0);  fadd_atomic(&Bs[r4.x], w4.x);
        fadd_atomic(&A[r4.y], w4.y * x1);  fadd_atomic(&Bs[r4.y], w4.y);
        fadd_atomic(&A[r4.z], w4.z * x2);  fadd_atomic(&Bs[r4.z], w4.z);
        fadd_atomic(&A[r4.w], w4.w * x3);  fadd_atomic(&Bs[r4.w], w4.w);
    } else {
        for (int e = base; e < E; ++e) {
            int r = rows[e], c = cols[e];
            float we = w[e];
            fadd_atomic(&A[r], we * x[c]);
            fadd_atomic(&Bs[r], we);
        }
    }
}

// Per-node MLP via WMMA: pre[m][n] = A[m]*W1[n] + B[m]*b1[n]  (16x4 @ 4x16, f32)
// then s2[m] = b2 + sum_n relu(pre[m][n]) * W2[n] over both 16-wide HID chunks.
__global__ void node_mlp_wmma(const float* __restrict__ A, const float* __restrict__ Bs,
                              const float* __restrict__ W1, const float* __restrict__ b1,
                              const float* __restrict__ W2, const float* __restrict__ b2,
                              float* __restrict__ s2, int n_nodes)
{
    int gid  = blockIdx.x * blockDim.x + threadIdx.x;
    int wave = gid >> 5;                 // wave32
    int lane = threadIdx.x & 31;
    int base = wave * 16;                // 16 nodes per wave
    int m    = lane & 15;
    bool lo  = lane < 16;

    // A-matrix 16x4 f32: lanes 0-15 hold K=0 (v0) / K=1 (v1); lanes 16-31 hold K=2/K=3 (zeros).
    int nidx = base + m; if (nidx >= n_nodes) nidx = n_nodes - 1;   // clamp, no branch
    float Av = A[nidx];
    float Bv = Bs[nidx];
    v2f a;
    a.x = lo ? Av : 0.0f;
    a.y = lo ? Bv : 0.0f;

    // B-matrix 4x16 f32: lane = N (mod 16); lanes 0-15 rows K=0 (W1) / K=1 (b1), upper half rows K=2/3 = 0.
    v2f bm0, bm1;
    bm0.x = lo ? W1[m]      : 0.0f;
    bm0.y = lo ? b1[m]      : 0.0f;
    bm1.x = lo ? W1[16 + m] : 0.0f;
    bm1.y = lo ? b1[16 + m] : 0.0f;

    float w2v0 = W2[m];
    float w2v1 = W2[16 + m];

    v8f c = {};
    // 8 args: (neg_a, A, neg_b, B, c_mod, C, reuse_a, reuse_b)
    v8f d0 = __builtin_amdgcn_wmma_f32_16x16x4_f32(false, a, false, bm0, (short)0, c, false, false);
    v8f d1 = __builtin_amdgcn_wmma_f32_16x16x4_f32(false, a, false, bm1, (short)0, c, false, false);

    // D layout: VGPR r holds row M=r (lanes 0-15) / M=r+8 (lanes 16-31), N = lane&15.
    float tot[8];
#pragma unroll
    for (int r = 0; r < 8; ++r) {
        float t0 = d0[r] > 0.0f ? d0[r] : 0.0f;
        float t1 = d1[r] > 0.0f ? d1[r] : 0.0f;
        float p  = t0 * w2v0 + t1 * w2v1;
        // reduce across the 16 lanes of each half (xor masks stay within halves)
        p += __shfl_xor(p, 1, 32);
        p += __shfl_xor(p, 2, 32);
        p += __shfl_xor(p, 4, 32);
        p += __shfl_xor(p, 8, 32);
        tot[r] = p;
    }

    float bias2 = b2[0];
#pragma unroll
    for (int r = 0; r < 8; ++r) {
        int out_node = base + r + (lo ? 0 : 8);
        if (m == r && out_node < n_nodes)
            s2[out_node] = tot[r] + bias2;
    }
}

// Pass 2: out[row] += w * s2[col].   4 edges per thread.
__global__ void edge_pass2(const int* __restrict__ rows, const int* __restrict__ cols,
                           const float* __restrict__ w, const float* __restrict__ s2,
                           float* __restrict__ out, int E)
{
    int t = blockIdx.x * blockDim.x + threadIdx.x;
    int base = t * 4;
    if (base + 4 <= E) {
        const int4   r4 = *(const int4*)(rows + base);
        const int4   c4 = *(const int4*)(cols + base);
        const float4 w4 = *(const float4*)(w + base);
        float s0 = s2[c4.x], s1 = s2[c4.y], s2v = s2[c4.z], s3 = s2[c4.w];
        fadd_atomic(&out[r4.x], w4.x * s0);
        fadd_atomic(&out[r4.y], w4.y * s1);
        fadd_atomic(&out[r4.z], w4.z * s2v);
        fadd_atomic(&out[r4.w], w4.w * s3);
    } else {
        for (int e = base; e < E; ++e) {
            int r = rows[e], c = cols[e];
            fadd_atomic(&out[r], w[e] * s2[c]);
        }
    }
}

extern "C" void kernel_launch(void* const* d_in, const int* in_sizes, int n_in,
                              void* d_out, int out_size, void* d_ws, size_t ws_size,
                              hipStream_t stream)
{
    // inputs: 0=t, 1=x[N], 2=edge_row[E], 3=edge_col[E], 4=edge_w[E], 5=W1[32], 6=b1[32], 7=W2[32], 8=b2[1]
    const float* x    = (const float*)d_in[1];
    const int*   erow = (const int*)d_in[2];
    const int*   ecol = (const int*)d_in[3];
    const float* ew   = (const float*)d_in[4];
    const float* W1   = (const float*)d_in[5];
    const float* b1   = (const float*)d_in[6];
    const float* W2   = (const float*)d_in[7];
    const float* b2   = (const float*)d_in[8];
    const int N = in_sizes[1];
    const int E = in_sizes[2];

    float* A   = (float*)d_ws;   // [N]
    float* Bs  = A + N;          // [N]
    float* s2  = Bs + N;         // [N]
    float* out = (float*)d_out;  // [N]

    hipMemsetAsync(A,   0, (size_t)2 * N * sizeof(float), stream);  // A and Bs
    hipMemsetAsync(out, 0, (size_t)N * sizeof(float),     stream);

    const int blk = 256;
    {
        int threads = (E + 3) / 4;
        edge_pass1<<<(threads + blk - 1) / blk, blk, 0, stream>>>(erow, ecol, ew, x, A, Bs, E);
    }
    {
        int waves   = (N + 15) / 16;
        int threads = waves * 32;
        node_mlp_wmma<<<(threads + blk - 1) / blk, blk, 0, stream>>>(A, Bs, W1, b1, W2, b2, s2, N);
    }
    {
        int threads = (E + 3) / 4;
        edge_pass2<<<(threads + blk - 1) / blk, blk, 0, stream>>>(erow, ecol, ew, s2, out, E);
    }
}